// GIN_20598663152197
// MI455X (gfx1250) — compile-verified
//
#include <hip/hip_runtime.h>

#define N_NODES 100000
#define N_EDGES 1600000
#define D 128
#define H 32
#define NTILES 6250          // N_NODES / 16
#define BN_EPS 1e-5f

typedef __attribute__((ext_vector_type(2))) float v2f;
typedef __attribute__((ext_vector_type(8))) float v8f;

__device__ __forceinline__ v8f wmma_f32(v2f a, v2f b, v8f c) {
  // D = A(16x4,f32) * B(4x16,f32) + C(16x16,f32)
  return __builtin_amdgcn_wmma_f32_16x16x4_f32(false, a, false, b, (short)0, c,
                                               false, false);
}

__device__ __forceinline__ v8f v8f_zero() {
  v8f z = {0.f, 0.f, 0.f, 0.f, 0.f, 0.f, 0.f, 0.f};
  return z;
}

// ---------------------------------------------------------------------------
// Kernel 1: Z = (1+eps[l]) * X  (also zeroes the per-layer stats buffer)
// ---------------------------------------------------------------------------
__global__ void gin_scale_init(float* __restrict__ Z, const float* __restrict__ X,
                               const float* __restrict__ epsP, int l,
                               float* __restrict__ stats) {
  if (blockIdx.x == 0 && threadIdx.x < 128) stats[threadIdx.x] = 0.0f;
  int t = blockIdx.x * 256 + threadIdx.x;      // N*D/4 threads exactly
  float s = 1.0f + epsP[l];
  float4 v = ((const float4*)X)[t];
  float4 o; o.x = v.x * s; o.y = v.y * s; o.z = v.z * s; o.w = v.w * s;
  ((float4*)Z)[t] = o;
}

// ---------------------------------------------------------------------------
// Kernel 2: Z[dst] += X[src] for every edge (hardware f32 L2 atomics).
// One thread handles one edge x 16 features.
// ---------------------------------------------------------------------------
__global__ void gin_scatter(float* __restrict__ Z, const float* __restrict__ X,
                            const long long* __restrict__ ei) {
  int t = blockIdx.x * 256 + threadIdx.x;      // E*8 threads exactly
  int e = t >> 3;
  int c = (t & 7) << 4;                        // 0,16,...,112
  long long s = ei[e];
  long long d = ei[N_EDGES + e];
  const float4* xs = (const float4*)(X + s * D + c);
  float* zd = Z + d * D + c;
#pragma unroll
  for (int i = 0; i < 4; ++i) {
    float4 v = xs[i];
    atomicAdd(zd + 4 * i + 0, v.x);
    atomicAdd(zd + 4 * i + 1, v.y);
    atomicAdd(zd + 4 * i + 2, v.z);
    atomicAdd(zd + 4 * i + 3, v.w);
  }
}

// ---------------------------------------------------------------------------
// Kernel 3: Y1 = Z @ W1^T + b1  (WMMA f32 16x16x4, K=128) + column stats
// W1 staged in LDS with padded row stride 132 (bank-conflict-free b64 reads).
// ---------------------------------------------------------------------------
__global__ void gin_gemm1(float* __restrict__ Y1, const float* __restrict__ Z,
                          const float* __restrict__ W1, const float* __restrict__ b1,
                          float* __restrict__ stats) {
  __shared__ float ldsW[32 * 132];
  __shared__ float ldsSum[32];
  __shared__ float ldsSq[32];
  int tid = threadIdx.x;
  for (int i = tid; i < 32 * 128; i += 256)
    ldsW[(i >> 7) * 132 + (i & 127)] = W1[i];
  if (tid < 32) { ldsSum[tid] = 0.f; ldsSq[tid] = 0.f; }
  __syncthreads();

  int lane = tid & 31, wave = tid >> 5;
  int mtile = blockIdx.x * 8 + wave;
  if (mtile < NTILES) {
    int laneM = lane & 15;
    int hi = lane >> 4;                        // 0: k,k+1   1: k+2,k+3
    const float* zrow = Z + (size_t)(mtile * 16 + laneM) * D + 2 * hi;
    const float* bw0 = ldsW + laneM * 132 + 2 * hi;
    const float* bw1 = ldsW + (16 + laneM) * 132 + 2 * hi;
    v8f acc0 = v8f_zero(), acc1 = v8f_zero();
#pragma unroll
    for (int kk = 0; kk < 32; ++kk) {
      int k0 = kk * 4;
      float2 az = *(const float2*)(zrow + k0);
      float2 b0z = *(const float2*)(bw0 + k0);
      float2 b1z = *(const float2*)(bw1 + k0);
      v2f a;  a.x = az.x;  a.y = az.y;
      v2f b0; b0.x = b0z.x; b0.y = b0z.y;
      v2f b1v; b1v.x = b1z.x; b1v.y = b1z.y;
      acc0 = wmma_f32(a, b0, acc0);
      acc1 = wmma_f32(a, b1v, acc1);
    }
    float bias0 = b1[laneM], bias1 = b1[16 + laneM];
    int rowbase = mtile * 16 + 8 * hi;
    float s0 = 0.f, q0 = 0.f, s1 = 0.f, q1 = 0.f;
#pragma unroll
    for (int r = 0; r < 8; ++r) {
      float y0 = acc0[r] + bias0;
      float y1 = acc1[r] + bias1;
      Y1[(size_t)(rowbase + r) * H + laneM] = y0;
      Y1[(size_t)(rowbase + r) * H + 16 + laneM] = y1;
      s0 += y0; q0 += y0 * y0;
      s1 += y1; q1 += y1 * y1;
    }
    atomicAdd(&ldsSum[laneM], s0);
    atomicAdd(&ldsSq[laneM], q0);
    atomicAdd(&ldsSum[16 + laneM], s1);
    atomicAdd(&ldsSq[16 + laneM], q1);
  }
  __syncthreads();
  if (tid < 32) {
    atomicAdd(&stats[tid], ldsSum[tid]);
    atomicAdd(&stats[32 + tid], ldsSq[tid]);
  }
}

// ---------------------------------------------------------------------------
// Kernel 4: h = relu(bn1(Y1)); Y2 = h @ W2^T + b2 (K=32) + column stats
// ---------------------------------------------------------------------------
__global__ void gin_gemm2(float* __restrict__ Y2, const float* __restrict__ Y1,
                          const float* __restrict__ W2, const float* __restrict__ b2,
                          const float* __restrict__ g1, const float* __restrict__ bt1,
                          float* __restrict__ stats) {
  __shared__ float ldsW[32 * 36];
  __shared__ float2 ldsAC[32];
  __shared__ float ldsSum[32];
  __shared__ float ldsSq[32];
  int tid = threadIdx.x;
  for (int i = tid; i < 32 * 32; i += 256)
    ldsW[(i >> 5) * 36 + (i & 31)] = W2[i];
  if (tid < 32) {
    float s = stats[tid], q = stats[32 + tid];
    float m = s * (1.0f / N_NODES);
    float v = q * (1.0f / N_NODES) - m * m;
    float a = g1[tid] * rsqrtf(v + BN_EPS);
    ldsAC[tid] = make_float2(a, bt1[tid] - a * m);
    ldsSum[tid] = 0.f; ldsSq[tid] = 0.f;
  }
  __syncthreads();

  int lane = tid & 31, wave = tid >> 5;
  int mtile = blockIdx.x * 8 + wave;
  if (mtile < NTILES) {
    int laneM = lane & 15;
    int hi = lane >> 4;
    const float* yrow = Y1 + (size_t)(mtile * 16 + laneM) * H + 2 * hi;
    const float* bw0 = ldsW + laneM * 36 + 2 * hi;
    const float* bw1 = ldsW + (16 + laneM) * 36 + 2 * hi;
    v8f acc0 = v8f_zero(), acc1 = v8f_zero();
#pragma unroll
    for (int kk = 0; kk < 8; ++kk) {
      int k0 = kk * 4;
      float2 yz = *(const float2*)(yrow + k0);
      float4 ac = *(const float4*)(&ldsAC[k0 + 2 * hi]);  // (a[k],c[k],a[k+1],c[k+1])
      v2f a;
      a.x = fmaxf(0.f, yz.x * ac.x + ac.y);
      a.y = fmaxf(0.f, yz.y * ac.z + ac.w);
      float2 b0z = *(const float2*)(bw0 + k0);
      float2 b1z = *(const float2*)(bw1 + k0);
      v2f b0; b0.x = b0z.x; b0.y = b0z.y;
      v2f b1v; b1v.x = b1z.x; b1v.y = b1z.y;
      acc0 = wmma_f32(a, b0, acc0);
      acc1 = wmma_f32(a, b1v, acc1);
    }
    float bias0 = b2[laneM], bias1 = b2[16 + laneM];
    int rowbase = mtile * 16 + 8 * hi;
    float s0 = 0.f, q0 = 0.f, s1 = 0.f, q1 = 0.f;
#pragma unroll
    for (int r = 0; r < 8; ++r) {
      float y0 = acc0[r] + bias0;
      float y1 = acc1[r] + bias1;
      Y2[(size_t)(rowbase + r) * H + laneM] = y0;
      Y2[(size_t)(rowbase + r) * H + 16 + laneM] = y1;
      s0 += y0; q0 += y0 * y0;
      s1 += y1; q1 += y1 * y1;
    }
    atomicAdd(&ldsSum[laneM], s0);
    atomicAdd(&ldsSq[laneM], q0);
    atomicAdd(&ldsSum[16 + laneM], s1);
    atomicAdd(&ldsSq[16 + laneM], q1);
  }
  __syncthreads();
  if (tid < 32) {
    atomicAdd(&stats[64 + tid], ldsSum[tid]);
    atomicAdd(&stats[96 + tid], ldsSq[tid]);
  }
}

// ---------------------------------------------------------------------------
// Kernel 5: h = relu(bn2(Y2)); Xout = h @ W3^T + b3 (K=32, 8 output N-tiles)
// ---------------------------------------------------------------------------
__global__ void gin_gemm3(float* __restrict__ Xout, const float* __restrict__ Y2,
                          const float* __restrict__ W3, const float* __restrict__ b3,
                          const float* __restrict__ g2, const float* __restrict__ bt2,
                          const float* __restrict__ stats) {
  __shared__ float ldsW[128 * 36];
  __shared__ float2 ldsAC[32];
  int tid = threadIdx.x;
  for (int i = tid; i < 128 * 32; i += 256)
    ldsW[(i >> 5) * 36 + (i & 31)] = W3[i];
  if (tid < 32) {
    float s = stats[64 + tid], q = stats[96 + tid];
    float m = s * (1.0f / N_NODES);
    float v = q * (1.0f / N_NODES) - m * m;
    float a = g2[tid] * rsqrtf(v + BN_EPS);
    ldsAC[tid] = make_float2(a, bt2[tid] - a * m);
  }
  __syncthreads();

  int lane = tid & 31, wave = tid >> 5;
  int mtile = blockIdx.x * 8 + wave;
  if (mtile >= NTILES) return;               // no barriers below
  int laneM = lane & 15;
  int hi = lane >> 4;
  const float* yrow = Y2 + (size_t)(mtile * 16 + laneM) * H + 2 * hi;
  v8f acc[8];
#pragma unroll
  for (int t2 = 0; t2 < 8; ++t2) acc[t2] = v8f_zero();
#pragma unroll
  for (int kk = 0; kk < 8; ++kk) {
    int k0 = kk * 4;
    float2 yz = *(const float2*)(yrow + k0);
    float4 ac = *(const float4*)(&ldsAC[k0 + 2 * hi]);
    v2f a;
    a.x = fmaxf(0.f, yz.x * ac.x + ac.y);
    a.y = fmaxf(0.f, yz.y * ac.z + ac.w);
#pragma unroll
    for (int t2 = 0; t2 < 8; ++t2) {
      float2 bz = *(const float2*)(ldsW + (t2 * 16 + laneM) * 36 + 2 * hi + k0);
      v2f b; b.x = bz.x; b.y = bz.y;
      acc[t2] = wmma_f32(a, b, acc[t2]);
    }
  }
  int rowbase = mtile * 16 + 8 * hi;
#pragma unroll
  for (int t2 = 0; t2 < 8; ++t2) {
    float bias = b3[t2 * 16 + laneM];
#pragma unroll
    for (int r = 0; r < 8; ++r)
      Xout[(size_t)(rowbase + r) * D + t2 * 16 + laneM] = acc[t2][r] + bias;
  }
}

// ---------------------------------------------------------------------------
extern "C" void kernel_launch(void* const* d_in, const int* in_sizes, int n_in,
                              void* d_out, int out_size, void* d_ws, size_t ws_size,
                              hipStream_t stream) {
  const float* X0   = (const float*)d_in[0];
  const long long* ei = (const long long*)d_in[1];   // int64 [2, E]
  const float* epsP = (const float*)d_in[2];
  const float* W1   = (const float*)d_in[3];
  const float* b1   = (const float*)d_in[4];
  const float* g1   = (const float*)d_in[5];
  const float* bt1  = (const float*)d_in[6];
  const float* W2   = (const float*)d_in[7];
  const float* b2   = (const float*)d_in[8];
  const float* g2   = (const float*)d_in[9];
  const float* bt2  = (const float*)d_in[10];
  const float* W3   = (const float*)d_in[11];
  const float* b3   = (const float*)d_in[12];
  float* out = (float*)d_out;

  float* Z     = (float*)d_ws;                       // N*128
  float* Y1    = Z + (size_t)N_NODES * D;            // N*32
  float* Y2    = Y1 + (size_t)N_NODES * H;           // N*32
  float* Xbuf  = Y2 + (size_t)N_NODES * H;           // N*128
  float* stats = Xbuf + (size_t)N_NODES * D;         // 128 floats

  const int gemmBlocks = (NTILES + 7) / 8;           // 782

  for (int l = 0; l < 4; ++l) {
    const float* Xin = (l == 0) ? X0 : Xbuf;
    float* Xout = (l == 3) ? out : Xbuf;
    gin_scale_init<<<(N_NODES * D / 4) / 256, 256, 0, stream>>>(Z, Xin, epsP, l, stats);
    gin_scatter<<<(N_EDGES * 8) / 256, 256, 0, stream>>>(Z, Xin, ei);
    gin_gemm1<<<gemmBlocks, 256, 0, stream>>>(Y1, Z, W1 + (size_t)l * H * D,
                                              b1 + l * H, stats);
    gin_gemm2<<<gemmBlocks, 256, 0, stream>>>(Y2, Y1, W2 + (size_t)l * H * H,
                                              b2 + l * H, g1 + l * H, bt1 + l * H, stats);
    gin_gemm3<<<gemmBlocks, 256, 0, stream>>>(Xout, Y2, W3 + (size_t)l * D * H,
                                              b3 + l * D, g2 + l * H, bt2 + l * H, stats);
  }
}